// DistanceLoss_21543555957021
// MI455X (gfx1250) — compile-verified
//
#include <hip/hip_runtime.h>
#include <hip/hip_bf16.h>
#include <stdint.h>

#define NIMG 16        // N*C = 8*2
#define NCH  2
#define H    512
#define W    512
#define TJ   32        // columns per pass-2 block
#define INF_D 1024.0f  // H+W, matches reference INF

// ---------------- CDNA5 async global->LDS support ----------------
#if defined(__has_builtin)
#if __has_builtin(__builtin_amdgcn_global_load_async_to_lds_b32)
#define HAVE_ASYNC_LDS 1
#endif
#endif

#if defined(HAVE_ASYNC_LDS)
typedef __attribute__((address_space(1))) int g_as_int;
typedef __attribute__((address_space(3))) int l_as_int;

__device__ __forceinline__ void async_copy_b32(const float* g, float* l) {
    __builtin_amdgcn_global_load_async_to_lds_b32(
        (g_as_int*)g,
        (l_as_int*)l,
        0, 0);
}
__device__ __forceinline__ void wait_async0() {
#if __has_builtin(__builtin_amdgcn_s_wait_asynccnt)
    __builtin_amdgcn_s_wait_asynccnt(0);
#else
    asm volatile("s_wait_asynccnt 0" ::: "memory");
#endif
}
#endif

// ---------------- Kernel 1: 1D row EDT via parallel prefix-min ----------------
// f[w] = w + min_{w'<=w}(g[w']-w'),  b[w] = -w + min_{w'>=w}(g[w']+w'), d1=min(f,b)
__global__ void __launch_bounds__(512)
edt_rows_kernel(const float* __restrict__ y_true, float* __restrict__ d1sq) {
    __shared__ float sA[W];
    __shared__ float sB[W];
    const int w   = threadIdx.x;
    const int row = blockIdx.x;                 // 0 .. NIMG*H-1
    const size_t base = (size_t)row * W;

    const float m = y_true[base + w];
    const float g = (m > 0.5f) ? 0.0f : INF_D;
    sA[w]             = g - (float)w;
    sB[(W - 1) - w]   = g + (float)w;
    __syncthreads();

    for (int off = 1; off < W; off <<= 1) {
        float a = sA[w], b = sB[w];
        if (w >= off) {
            a = fminf(a, sA[w - off]);
            b = fminf(b, sB[w - off]);
        }
        __syncthreads();
        sA[w] = a; sB[w] = b;
        __syncthreads();
    }

    const float f  = (float)w + sA[w];
    const float bk = -(float)w + sB[(W - 1) - w];
    const float d  = fminf(f, bk);
    d1sq[base + w] = d * d;                     // exact: d <= 1024, d^2 < 2^24
}

// ---------------- Kernel 2: column lower-envelope + fused loss ----------------
// One block = 32 columns of one (n,c) image. 512 threads: t = ib*32 + jj,
// ib in [0,16), jj in [0,32). Thread handles rows i = ib + 16*r, r in [0,32).
__global__ void __launch_bounds__(512)
edt_cols_loss_kernel(const float* __restrict__ y_pred,
                     const float* __restrict__ y_true,
                     const float* __restrict__ d1sq,
                     float* __restrict__ partial) {
    __shared__ float tile[H * TJ];   // 64 KB: tile[k*TJ + jj] = d1sq[k][j0+jj]
    __shared__ float red[512];

    const int t  = threadIdx.x;
    const int jj = t & (TJ - 1);
    const int ib = t >> 5;                       // 0..15
    const int blk = blockIdx.x;                  // 0 .. NIMG*(W/TJ)-1
    const int nc  = blk / (W / TJ);
    const int j0  = (blk % (W / TJ)) * TJ;
    const int n   = nc >> 1;
    const int c   = nc & 1;
    const size_t imgBase = (size_t)nc * H * W;
    const size_t othBase = (size_t)(n * NCH + (c ^ 1)) * H * W;

    // Stage the 512x32 d1sq tile into LDS (coalesced rows; async on CDNA5)
    for (int k = ib; k < H; k += 16) {
        const float* gp = d1sq + imgBase + (size_t)k * W + (j0 + jj);
#if defined(HAVE_ASYNC_LDS)
        async_copy_b32(gp, &tile[k * TJ + jj]);
#else
        tile[k * TJ + jj] = *gp;
#endif
    }
#if defined(HAVE_ASYNC_LDS)
    wait_async0();
#endif
    __syncthreads();

    // 32 running minima per thread: each LDS value read once per wave step
    float best[32];
#pragma unroll
    for (int r = 0; r < 32; ++r) best[r] = 3.0e38f;

    for (int k = 0; k < H; ++k) {
        const float val = tile[k * TJ + jj];     // wave reads 32 consecutive DWORDs
        const float dk0 = (float)(ib - k);
#pragma unroll
        for (int r = 0; r < 32; ++r) {
            const float d = dk0 + (float)(16 * r);
            best[r] = fminf(best[r], __fmaf_rn(d, d, val));
        }
    }

    // Fused softmax / sqe / (1+dm) loss for this thread's 32 pixels
    float acc = 0.0f;
#pragma unroll
    for (int r = 0; r < 32; ++r) {
        const int i = ib + 16 * r;
        const float dm = sqrtf(best[r]) * (1.0f / 511.0f);
        const size_t px = (size_t)i * W + (j0 + jj);
        const float xc = y_pred[imgBase + px];
        const float xo = y_pred[othBase + px];
        const float p  = 1.0f / (1.0f + expf(xo - xc));
        const float tv = y_true[imgBase + px];
        const float e  = p - tv;
        acc += (1.0f + dm) * (e * e);
    }

    // Deterministic block tree reduction
    red[t] = acc;
    __syncthreads();
    for (int s = 256; s > 0; s >>= 1) {
        if (t < s) red[t] += red[t + s];
        __syncthreads();
    }
    if (t == 0) partial[blk] = red[0];
}

// ---------------- Kernel 3: final reduce (WMMA dot-with-ones) ----------------
typedef __attribute__((ext_vector_type(2))) float v2f;
typedef __attribute__((ext_vector_type(8))) float v8f;

__global__ void __launch_bounds__(256)
final_reduce_kernel(const float* __restrict__ partial, float* __restrict__ out,
                    int npart) {
    __shared__ float s[256];
    const int t = threadIdx.x;
    float a = 0.0f;
    for (int idx = t; idx < npart; idx += 256) a += partial[idx];
    s[t] = a;
    __syncthreads();

    if (t < 32) {   // wave 0, EXEC all ones (WMMA requirement)
#if defined(__has_builtin) && __has_builtin(__builtin_amdgcn_wmma_f32_16x16x4_f32)
        v8f accv = {};
        v2f bv;  bv[0] = 1.0f; bv[1] = 1.0f;          // B = ones -> layout-agnostic row sums
#pragma unroll
        for (int chunk = 0; chunk < 256; chunk += 64) {
            v2f av;
            av[0] = s[chunk + t];
            av[1] = s[chunk + 32 + t];
            accv = __builtin_amdgcn_wmma_f32_16x16x4_f32(
                false, av, false, bv, (short)0, accv, false, false);
        }
        float sum8 = accv[0] + accv[1] + accv[2] + accv[3] +
                     accv[4] + accv[5] + accv[6] + accv[7];
        // D[i,j] identical over j: total = rows 0..7 (lane 0) + rows 8..15 (lane 16)
        const float tot = __shfl(sum8, 0, 32) + __shfl(sum8, 16, 32);
        if (t == 0) out[0] = tot * (1.0f / (float)((size_t)NIMG * H * W));
#else
        if (t == 0) {
            float tot = 0.0f;
            for (int i = 0; i < 256; ++i) tot += s[i];
            out[0] = tot * (1.0f / (float)((size_t)NIMG * H * W));
        }
#endif
    }
}

// ---------------- Host-side launch ----------------
extern "C" void kernel_launch(void* const* d_in, const int* in_sizes, int n_in,
                              void* d_out, int out_size, void* d_ws, size_t ws_size,
                              hipStream_t stream) {
    (void)in_sizes; (void)n_in; (void)out_size; (void)ws_size;
    const float* y_pred = (const float*)d_in[0];
    const float* y_true = (const float*)d_in[1];

    float* d1sq    = (float*)d_ws;                        // NIMG*H*W floats = 16 MB
    float* partial = d1sq + (size_t)NIMG * H * W;         // 256 floats

    edt_rows_kernel<<<NIMG * H, 512, 0, stream>>>(y_true, d1sq);
    edt_cols_loss_kernel<<<NIMG * (W / TJ), 512, 0, stream>>>(y_pred, y_true,
                                                              d1sq, partial);
    final_reduce_kernel<<<1, 256, 0, stream>>>(partial, (float*)d_out,
                                               NIMG * (W / TJ));
}